// PureTriadicBrain_73753178407499
// MI455X (gfx1250) — compile-verified
//
#include <hip/hip_runtime.h>
#include <math.h>

typedef float v2f __attribute__((ext_vector_type(2)));
typedef float v4f __attribute__((ext_vector_type(4)));
typedef float v8f __attribute__((ext_vector_type(8)));

#define NB 8      // batch
#define NN 64     // nodes
#define ND 256    // node dim
#define NH 64     // heads (head_dim = 4)
#define SP 65     // LDS row stride for 64x64 score matrix (bank-conflict pad)

__device__ __forceinline__ float buntanh_f(float x) {
    // (tanh(x) + 0.6 x) * sqrt(2 + sqrt(2))
    return (tanhf(x) + 0.6f * x) * 1.8477590650225735f;
}

__global__ void zero_out_kernel(float* __restrict__ out) { out[0] = 0.0f; }

// ---------------------------------------------------------------------------
// Phase 1: per-node GEMV  y = state[b,n,:] @ W[b,n,:,:], then buntanh.
// Pure HBM-stream kernel: 402 MB of weights read once -> non-temporal b128
// loads, fully coalesced rows, 4-way row-group split reduced through LDS.
// ---------------------------------------------------------------------------
__global__ __launch_bounds__(256)
void gemv_buntanh_kernel(const float* __restrict__ state,
                         const float* __restrict__ w1,
                         const float* __restrict__ w2,
                         const float* __restrict__ w3,
                         float* __restrict__ pred,
                         float* __restrict__ keys,
                         float* __restrict__ quer)
{
    const int node  = blockIdx.x;        // b*64 + n
    const int which = blockIdx.y;        // 0:w1->pred 1:w2->keys 2:w3->quer
    const float* __restrict__ W = (which == 0) ? w1 : (which == 1) ? w2 : w3;
    float* __restrict__ out     = (which == 0) ? pred : (which == 1) ? keys : quer;

    __shared__ float xs[ND];
    __shared__ v4f   part[4][64];

    const int tid = threadIdx.x;
    xs[tid] = state[(size_t)node * ND + tid];
    __syncthreads();

    const int cg = tid & 63;   // column group: owns columns 4*cg .. 4*cg+3
    const int rg = tid >> 6;   // row group 0..3: rows rg, rg+4, rg+8, ...
    const v4f* __restrict__ Wv = (const v4f*)(W + (size_t)node * ND * ND);

    v4f acc = {0.0f, 0.0f, 0.0f, 0.0f};
    #pragma unroll 4
    for (int j = rg; j < ND; j += 4) {
        if (j + 32 < ND)  // lookahead prefetch (global_prefetch_b8)
            __builtin_prefetch(&Wv[(size_t)(j + 32) * 64 + cg], 0, 0);
        // streamed-once weights: non-temporal 128-bit load
        v4f w = __builtin_nontemporal_load(&Wv[(size_t)j * 64 + cg]);
        acc += w * xs[j];
    }
    part[rg][cg] = acc;
    __syncthreads();

    if (rg == 0) {
        v4f t = part[0][cg] + part[1][cg] + part[2][cg] + part[3][cg];
        v4f r;
        r[0] = buntanh_f(t[0]);
        r[1] = buntanh_f(t[1]);
        r[2] = buntanh_f(t[2]);
        r[3] = buntanh_f(t[3]);
        ((v4f*)(out + (size_t)node * ND))[cg] = r;
    }
}

// ---------------------------------------------------------------------------
// Phase 2: one wave32 per (b,h).
//   S = (0.5*Q) K^T   via 16x V_WMMA_F32_16X16X4_F32 (HEAD_DIM==4 == WMMA K)
//   rms-norm stats harvested from WMMA accumulators, Sinkhorn x5 in LDS,
//   threshold, routed_V = A @ V, squared-error loss -> atomicAdd(scalar).
// Q/K staged as scalar float LDS so operand pairs load as aligned ds_load_b64
// (avoids b128 + cndmask component selection).
// ---------------------------------------------------------------------------
__global__ __launch_bounds__(32)
void attn_sinkhorn_loss_kernel(const float* __restrict__ pred,
                               const float* __restrict__ keys,
                               const float* __restrict__ quer,
                               const float* __restrict__ env,    // [8,1024]
                               const float* __restrict__ state,  // [8,64,256]
                               float* __restrict__ out)
{
    const int bh = blockIdx.x;       // 0..511
    const int b  = bh >> 6;
    const int h  = bh & 63;

    __shared__ float S[NN * SP];     // 64 x 64 scores (stride 65)
    __shared__ float Qs[NN * 4];     // 0.5 * Q   (scale folded into A operand)
    __shared__ float Ks[NN * 4];
    __shared__ v4f   Vs[NN];

    const int l = threadIdx.x;       // 0..31, EXEC all ones (WMMA requirement)

    #pragma unroll
    for (int r = 0; r < 2; ++r) {
        const int n = l + r * 32;
        const size_t base = ((size_t)(b * NN + n)) * ND + h * 4;
        v4f q = *(const v4f*)(quer + base);   // queries <- w3 branch
        *(v4f*)&Qs[n * 4] = q * 0.5f;         // fold 1/sqrt(HEAD_DIM)
        *(v4f*)&Ks[n * 4] = *(const v4f*)(keys + base);  // keys <- w2 branch
        Vs[n] = *(const v4f*)(pred + base);   // values <- w1 branch
    }
    __syncthreads();

    const int lr = l & 15;   // M/N row within 16-tile
    const int lh = l >> 4;   // which half of the wave (selects K pair 0,1|2,3)

    float sumsq = 0.0f;
    float mx    = -3.4e38f;

    // S[m][n] = sum_k (0.5*Q[m][k]) * K[n][k], 4x4 tiles of 16x16 WMMA output
    #pragma unroll
    for (int mt = 0; mt < 4; ++mt) {
        // A 16x4 f32: lanes 0-15 hold K=0,1; lanes 16-31 hold K=2,3
        v2f a = *(const v2f*)&Qs[(mt * 16 + lr) * 4 + lh * 2];
        #pragma unroll
        for (int nt = 0; nt < 4; ++nt) {
            // B 4x16 f32: B[k][n] = K[n][k], same lane pattern
            v2f bb = *(const v2f*)&Ks[(nt * 16 + lr) * 4 + lh * 2];
            v8f c = {};
            c = __builtin_amdgcn_wmma_f32_16x16x4_f32(
                    false, a, false, bb, (short)0, c, false, false);
            #pragma unroll
            for (int r = 0; r < 8; ++r) {
                float s = c[r];
                S[(mt * 16 + lh * 8 + r) * SP + (nt * 16 + lr)] = s;
                sumsq += s * s;
                mx = fmaxf(mx, s);
            }
        }
    }
    __syncthreads();

    // wave reduction of rms stats + max (for stabilized exp)
    #pragma unroll
    for (int off = 16; off > 0; off >>= 1) {
        sumsq += __shfl_xor(sumsq, off, 32);
        mx = fmaxf(mx, __shfl_xor(mx, off, 32));
    }
    const float scale = rsqrtf(sumsq * (1.0f / 4096.0f) + 1e-8f);

    // p = exp(norm(S) - max(norm(S))) = exp((S - mx) * scale)   (scale > 0)
    for (int r = 0; r < 2; ++r) {
        const int n = l + r * 32;
        for (int m = 0; m < NN; ++m) {
            float v = S[n * SP + m];
            S[n * SP + m] = expf((v - mx) * scale);
        }
    }
    __syncthreads();

    // Sinkhorn-Knopp: 5 x (row normalize, then column normalize)
    for (int it = 0; it < 5; ++it) {
        for (int r = 0; r < 2; ++r) {
            const int n = l + r * 32;
            float s = 0.0f;
            for (int m = 0; m < NN; ++m) s += S[n * SP + m];
            const float inv = 1.0f / (s + 1e-8f);
            for (int m = 0; m < NN; ++m) S[n * SP + m] *= inv;
        }
        __syncthreads();
        for (int r = 0; r < 2; ++r) {
            const int m = l + r * 32;  // stride-65 columns -> conflict free
            float s = 0.0f;
            for (int n = 0; n < NN; ++n) s += S[n * SP + m];
            const float inv = 1.0f / (s + 1e-8f);
            for (int n = 0; n < NN; ++n) S[n * SP + m] *= inv;
        }
        __syncthreads();
    }

    // threshold, routed_V = A @ V, loss contribution for this (b,h)
    const float thr = 1.0f / 64.0f;
    float env_sum = 0.0f, mem_sum = 0.0f;
    for (int r = 0; r < 2; ++r) {
        const int n = l + r * 32;
        v4f acc = {0.0f, 0.0f, 0.0f, 0.0f};
        for (int m = 0; m < NN; ++m) {
            float p = S[n * SP + m];
            p = (p > thr) ? p : 0.0f;
            acc += Vs[m] * p;
        }
        if (n < 4) {  // env nodes: target = env_input.reshape(B,4,256)
            const v4f e = *(const v4f*)(env + (size_t)b * 1024 + (size_t)n * 256 + h * 4);
            v4f d = acc - e;
            env_sum += d[0] * d[0] + d[1] * d[1] + d[2] * d[2] + d[3] * d[3];
        } else {      // memory nodes: target = state
            const v4f e = *(const v4f*)(state + ((size_t)(b * NN + n)) * ND + h * 4);
            v4f d = acc - e;
            mem_sum += d[0] * d[0] + d[1] * d[1] + d[2] * d[2] + d[3] * d[3];
        }
    }
    #pragma unroll
    for (int off = 16; off > 0; off >>= 1) {
        env_sum += __shfl_xor(env_sum, off, 32);
        mem_sum += __shfl_xor(mem_sum, off, 32);
    }
    if (l == 0)
        atomicAdd(out, env_sum * (1.0f / 8192.0f)      // mean over B*4*256
                     + mem_sum * (0.5f / 122880.0f));  // 0.5 * mean over B*60*256
}

// ---------------------------------------------------------------------------
extern "C" void kernel_launch(void* const* d_in, const int* in_sizes, int n_in,
                              void* d_out, int out_size, void* d_ws, size_t ws_size,
                              hipStream_t stream) {
    (void)in_sizes; (void)n_in; (void)out_size; (void)ws_size;
    const float* env   = (const float*)d_in[0];  // [8,1024]
    const float* w1    = (const float*)d_in[1];  // [8,64,256,256]
    const float* w2    = (const float*)d_in[2];
    const float* w3    = (const float*)d_in[3];
    const float* state = (const float*)d_in[4];  // [8,64,256]
    float* out = (float*)d_out;

    float* pred = (float*)d_ws;                       // 512 KB each
    float* keys = pred + (size_t)NB * NN * ND;
    float* quer = keys + (size_t)NB * NN * ND;

    zero_out_kernel<<<1, 1, 0, stream>>>(out);

    dim3 g1(NB * NN, 3);
    gemv_buntanh_kernel<<<g1, 256, 0, stream>>>(state, w1, w2, w3,
                                                pred, keys, quer);

    attn_sinkhorn_loss_kernel<<<NB * NH, 32, 0, stream>>>(pred, keys, quer,
                                                          env, state, out);
}